// GHMLoss_20383914787641
// MI455X (gfx1250) — compile-verified
//
#include <hip/hip_runtime.h>

#define BINS 10
#define NCLS 16

typedef __attribute__((ext_vector_type(2))) float v2f;
typedef __attribute__((ext_vector_type(8))) float v8f;

#define L2E 1.4426950408889634f   // log2(e)
#define LN2 0.6931471805599453f   // ln(2)

// Native transcendental pipe (v_exp_f32 / v_log_f32 / v_rcp_f32).
// AMDGCN builtins only exist in the device pass, so gate on
// __HIP_DEVICE_COMPILE__; fallbacks resolve in both passes.
static __device__ __forceinline__ float fast_exp2(float x) {
#if defined(__HIP_DEVICE_COMPILE__) && __has_builtin(__builtin_amdgcn_exp2f)
    return __builtin_amdgcn_exp2f(x);
#else
    return exp2f(x);
#endif
}
static __device__ __forceinline__ float fast_log2(float x) {
#if defined(__HIP_DEVICE_COMPILE__) && __has_builtin(__builtin_amdgcn_logf)
    return __builtin_amdgcn_logf(x);
#else
    return log2f(x);
#endif
}
static __device__ __forceinline__ float fast_rcp(float x) {
#if defined(__HIP_DEVICE_COMPILE__) && __has_builtin(__builtin_amdgcn_rcpf)
    return __builtin_amdgcn_rcpf(x);
#else
    return 1.0f / x;
#endif
}

// ---------------------------------------------------------------------------
// Kernel 1: zero the 20-float accumulator in workspace (counts[10], nll[10]).
// ---------------------------------------------------------------------------
__global__ void __launch_bounds__(32) ghm_zero(float* __restrict__ acc) {
    int t = threadIdx.x;
    if (t < 2 * BINS) acc[t] = 0.0f;
}

// ---------------------------------------------------------------------------
// Kernel 2: streaming pass. Each wave processes 16-row tiles.
//   - lane L<16 loads even float-pairs of row L, lane L+16 the odd pairs
//     (exactly the 16x4 f32 A-matrix layout)
//   - target + target-logit loads hoisted early (latency hidden by exp/WMMA)
//   - row max via fmax + shfl_xor(16)
//   - base-2 softmax: one v_fma + one v_exp per element (TRANS co-exec)
//   - Sum(exp) via 4 accumulating V_WMMA_F32_16X16X4_F32 with B = ones
//   - 16 stat lanes compute nll / p_t / bin, accumulate in LDS (ds_add_f32)
//   - one global atomic merge per block per bin
// ---------------------------------------------------------------------------
__global__ void __launch_bounds__(256) ghm_main(const float* __restrict__ in,
                                                const int* __restrict__ tgt,
                                                float* __restrict__ acc,
                                                int nTiles) {
    __shared__ float sb[2 * BINS];
    const int tid = threadIdx.x;
    if (tid < 2 * BINS) sb[tid] = 0.0f;
    __syncthreads();

    const int lane    = tid & 31;
    const int wib     = tid >> 5;
    const int wpb     = blockDim.x >> 5;
    const int wave    = blockIdx.x * wpb + wib;
    const int wstride = gridDim.x * wpb;

    const int row  = lane & 15;   // row inside 16-row tile
    const int half = lane >> 4;   // 0: cols {4j,4j+1}, 1: cols {4j+2,4j+3}

    const v2f ones = {1.0f, 1.0f};

    for (int tile = wave; tile < nTiles; tile += wstride) {
        const float* base = in + (size_t)tile * (16 * NCLS);
        const float* rp   = base + row * NCLS + 2 * half;

        v2f a0 = *(const v2f*)(rp + 0);
        v2f a1 = *(const v2f*)(rp + 4);
        v2f a2 = *(const v2f*)(rp + 8);
        v2f a3 = *(const v2f*)(rp + 12);

        // Hoisted: target class + target logit for this lane's row (all 32
        // lanes issue; lanes L and L+16 fetch the same address; the xt load
        // hits the just-fetched tile in near cache). Latency is hidden by
        // the exp/WMMA section below instead of stalling the stat section.
        const int   t  = tgt[tile * 16 + row];   // target class (0 or 1)
        const float xt = base[row * NCLS + t];

        if (tile + wstride < nTiles) {
            __builtin_prefetch(in + (size_t)(tile + wstride) * (16 * NCLS) +
                               row * NCLS + 2 * half, 0, 0);
        }

        // Row max: partial over this lane's 8 values, then swap halves.
        float m = fmaxf(fmaxf(fmaxf(a0[0], a0[1]), fmaxf(a1[0], a1[1])),
                        fmaxf(fmaxf(a2[0], a2[1]), fmaxf(a3[0], a3[1])));
        m = fmaxf(m, __shfl_xor(m, 16, 32));

        // E = exp2(x*log2e - m*log2e): one fma + one v_exp per element.
        const float mb = m * L2E;
        a0[0] = fast_exp2(fmaf(a0[0], L2E, -mb));
        a0[1] = fast_exp2(fmaf(a0[1], L2E, -mb));
        a1[0] = fast_exp2(fmaf(a1[0], L2E, -mb));
        a1[1] = fast_exp2(fmaf(a1[1], L2E, -mb));
        a2[0] = fast_exp2(fmaf(a2[0], L2E, -mb));
        a2[1] = fast_exp2(fmaf(a2[1], L2E, -mb));
        a3[0] = fast_exp2(fmaf(a3[0], L2E, -mb));
        a3[1] = fast_exp2(fmaf(a3[1], L2E, -mb));

        // D = E * ones : row sums replicated across all 16 columns.
        v8f d = {};
        d = __builtin_amdgcn_wmma_f32_16x16x4_f32(false, a0, false, ones, (short)0, d, false, false);
        d = __builtin_amdgcn_wmma_f32_16x16x4_f32(false, a1, false, ones, (short)0, d, false, false);
        d = __builtin_amdgcn_wmma_f32_16x16x4_f32(false, a2, false, ones, (short)0, d, false, false);
        d = __builtin_amdgcn_wmma_f32_16x16x4_f32(false, a3, false, ones, (short)0, d, false, false);

        // D layout: VGPR k holds S[k] on lanes 0-15, S[k+8] on lanes 16-31.
        // Stat lanes: rows 0-7 -> lanes 0-7 (S at d[row]);
        //             rows 8-15 -> lanes 24-31 (S at d[row-8]); both idx = lane&7.
        if ((lane < 8) || (lane >= 24)) {
            const int k = lane & 7;
            float S = d[0];
            S = (k == 1) ? d[1] : S;
            S = (k == 2) ? d[2] : S;
            S = (k == 3) ? d[3] : S;
            S = (k == 4) ? d[4] : S;
            S = (k == 5) ? d[5] : S;
            S = (k == 6) ? d[6] : S;
            S = (k == 7) ? d[7] : S;

            // nll = log(S) + m - xt    (log via v_log_f32 * ln2)
            const float nll = fmaf(LN2, fast_log2(S), m - xt);
            // p_t = exp(xt - m) / S    (v_exp + v_rcp)
            const float pt  = fast_exp2((xt - m) * L2E) * fast_rcp(S);
            const float g   = fabsf(pt - (float)t);

            // bin = count of edges (0.1..1.0) <= g, clipped to 9
            int bin = 0;
            bin += (g >= 0.1f); bin += (g >= 0.2f); bin += (g >= 0.3f);
            bin += (g >= 0.4f); bin += (g >= 0.5f); bin += (g >= 0.6f);
            bin += (g >= 0.7f); bin += (g >= 0.8f); bin += (g >= 0.9f);
            bin += (g >= 1.0f);
            if (bin > BINS - 1) bin = BINS - 1;

            atomicAdd(&sb[bin], 1.0f);
            atomicAdd(&sb[BINS + bin], nll);
        }
    }

    __syncthreads();
    if (tid < 2 * BINS) atomicAdd(&acc[tid], sb[tid]);
}

// ---------------------------------------------------------------------------
// Kernel 3: finalize. total = sum_b nllsum[b] * (N/10)/max(count[b],1)
// ---------------------------------------------------------------------------
__global__ void __launch_bounds__(32) ghm_fin(const float* __restrict__ acc,
                                              float* __restrict__ out, float Nf) {
    if (threadIdx.x == 0) {
        float total = 0.0f;
#pragma unroll
        for (int b = 0; b < BINS; ++b) {
            float cnt = fmaxf(acc[b], 1.0f);
            total += acc[BINS + b] * ((Nf / (float)BINS) / cnt);
        }
        *out = total;
    }
}

// ---------------------------------------------------------------------------
extern "C" void kernel_launch(void* const* d_in, const int* in_sizes, int n_in,
                              void* d_out, int out_size, void* d_ws, size_t ws_size,
                              hipStream_t stream) {
    const float* in  = (const float*)d_in[0];
    const int*   tgt = (const int*)d_in[1];
    float*       out = (float*)d_out;
    float*       acc = (float*)d_ws;

    const int N      = in_sizes[1];   // number of rows (target count)
    const int nTiles = N / 16;

    ghm_zero<<<1, 32, 0, stream>>>(acc);

    int blocks = 2048;                 // 8 waves/block -> 16384 waves, ~16 tiles each
    if (blocks > nTiles) blocks = (nTiles > 0) ? nTiles : 1;
    ghm_main<<<blocks, 256, 0, stream>>>(in, tgt, acc, nTiles);

    ghm_fin<<<1, 32, 0, stream>>>(acc, out, (float)N);
}